// HiddenMarkovGFormulaV2_43550968381855
// MI455X (gfx1250) — compile-verified
//
#include <hip/hip_runtime.h>
#include <stdint.h>

#define TT      64      // timesteps (reference constant)
#define ROWS    64      // subjects per workgroup
#define STRIDE  66      // padded LDS row stride (words): 66*tid mod 64 = 2*tid -> conflict-free wave32
#define THREADS 64

#if defined(__gfx1250__) && __has_builtin(__builtin_amdgcn_global_load_async_to_lds_b64) && __has_builtin(__builtin_amdgcn_global_store_async_from_lds_b64)
#define USE_ASYNC 1
#else
#define USE_ASYNC 0
#endif

#if USE_ASYNC
typedef int v2i __attribute__((ext_vector_type(2)));
typedef __attribute__((address_space(1))) v2i as1_v2i;   // global
typedef __attribute__((address_space(3))) v2i as3_v2i;   // LDS

__device__ __forceinline__ as1_v2i* to_gbl(const void* p) {
  // global flat address == addrspace(1) address
  return (as1_v2i*)(uintptr_t)p;
}
__device__ __forceinline__ as3_v2i* to_lds(const void* p) {
  // low 32 bits of a flat LDS pointer are the LDS byte offset (ISA 10.2 aperture rule)
  return (as3_v2i*)(uint32_t)(uintptr_t)p;
}
__device__ __forceinline__ void wait_async0() {
#if __has_builtin(__builtin_amdgcn_s_wait_asynccnt)
  __builtin_amdgcn_s_wait_asynccnt(0);
#else
  asm volatile("s_wait_asynccnt 0" ::: "memory");
#endif
}
#endif

// copy one b64 chunk (2 floats) global -> padded LDS tile
__device__ __forceinline__ void ld_chunk(const float* g, float* l) {
#if USE_ASYNC
  __builtin_amdgcn_global_load_async_to_lds_b64(to_gbl(g), to_lds(l), 0, 0);
#else
  *(float2*)l = *(const float2*)g;
#endif
}
// copy one b64 chunk padded LDS tile -> global
__device__ __forceinline__ void st_chunk(float* g, const float* l) {
#if USE_ASYNC
  __builtin_amdgcn_global_store_async_from_lds_b64(to_gbl(g), to_lds(l), 0, 0);
#else
  *(float2*)g = *(const float2*)l;
#endif
}

__global__ __launch_bounds__(THREADS) void hmm_filter_kernel(
    const float* __restrict__ G, const float* __restrict__ S,
    const float* __restrict__ C, const float* __restrict__ Y,
    const float* __restrict__ L,
    const float* psi_p, const float* gS_p, const float* gC_p, const float* gG_p,
    const float* gGS_p, const float* gGC_p, const float* gLw_p,
    const float* lsz_p, const float* b0_p, const float* bZ_p, const float* bS_p,
    const float* bG_p, const float* bGS_p, const float* bLw_p,
    float* __restrict__ Zf, float* __restrict__ Zvar, float* __restrict__ ws,
    int N, int K)
{
  __shared__ float tS[ROWS * STRIDE];
  __shared__ float tC[ROWS * STRIDE];
  __shared__ float tY[ROWS * STRIDE];
  __shared__ float red[THREADS];

  const int tid  = threadIdx.x;
  const int n0   = blockIdx.x * ROWS;
  const bool full = (n0 + ROWS) <= N;     // true for every block when N % ROWS == 0

  // ---------- stage inputs: contiguous global tile -> padded LDS tiles ----------
  // Tile = ROWS rows of TT floats (pitch TT). b64 chunks: 32 per row.
  // chunk c: row = c>>5, j = c&31; global float off = n0*TT + 2*c; lds word off = row*STRIDE + 2*j.
  const int PER_THREAD = (ROWS * (TT / 2)) / THREADS;   // 32
  if (full) {
#pragma unroll 8
    for (int i = 0; i < PER_THREAD; ++i) {
      const int c   = i * THREADS + tid;                // lanes consecutive -> coalesced
      const size_t goff = (size_t)n0 * TT + (size_t)c * 2;
      const int    loff = (c >> 5) * STRIDE + 2 * (c & 31);
      ld_chunk(S + goff, tS + loff);
      ld_chunk(C + goff, tC + loff);
      ld_chunk(Y + goff, tY + loff);
    }
  } else {
    for (int i = 0; i < PER_THREAD; ++i) {
      const int c   = i * THREADS + tid;
      const int row = c >> 5;
      if (n0 + row < N) {
        const size_t goff = (size_t)n0 * TT + (size_t)c * 2;
        const int    loff = row * STRIDE + 2 * (c & 31);
        ld_chunk(S + goff, tS + loff);
        ld_chunk(C + goff, tC + loff);
        ld_chunk(Y + goff, tY + loff);
      }
    }
  }
#if USE_ASYNC
  wait_async0();
#endif
  __syncthreads();

  // ---------- uniform parameters ----------
  const float psi  = psi_p[0];
  const float gSc  = gS_p[0],  gCc  = gC_p[0],  gGc  = gG_p[0];
  const float gGSc = gGS_p[0], gGCc = gGC_p[0];
  const float b0   = b0_p[0],  bZ   = bZ_p[0],  bSc  = bS_p[0];
  const float bGc  = bG_p[0],  bGSc = bGS_p[0];
  const float sigma2 = __expf(2.0f * lsz_p[0]);   // (e^{log_sigma})^2
  const float psi2 = psi * psi;
  const float bZ2  = bZ * bZ;

  // ---------- per-subject sequential filter ----------
  const int n = n0 + tid;
  float llacc = 0.0f;
  if (n < N) {
    const float Gv = G[n];
    float gl = 0.0f, bl = 0.0f;
    for (int k = 0; k < K; ++k) {
      const float lv = L[(size_t)n * K + k];
      gl = fmaf(lv, gLw_p[k], gl);
      bl = fmaf(lv, bLw_p[k], bl);
    }
    const float cg = gGc * Gv + gl;        // transition constant term
    const float cb = b0 + bGc * Gv + bl;   // outcome constant term

    float* __restrict__ rS = tS + tid * STRIDE;
    float* __restrict__ rC = tC + tid * STRIDE;
    float* __restrict__ rY = tY + tid * STRIDE;

    float Zm = 0.0f, Zv = 1.0f;
#pragma unroll 4
    for (int t = 0; t < TT; ++t) {
      const float St = rS[t];
      const float Ct = rC[t];
      const float Yt = rY[t];
      const float GS = Gv * St;
      const float GC = Gv * Ct;

      const float Zpred = psi * Zm + gSc * St + gCc * Ct + gGSc * GS + gGCc * GC + cg;
      const float Zpv   = psi2 * Zv + sigma2;

      float logit = cb + bZ * Zpred + bSc * St + bGSc * GS;
      logit = fminf(fmaxf(logit, -20.0f), 20.0f);
      const float p = 1.0f / (1.0f + __expf(-logit));

      const float grad = (Yt - p) * bZ;
      const float hess = bZ2 * p * (1.0f - p) + 1e-6f;
      const float Zpost_v = Zpv / (1.0f + Zpv * hess);   // == 1/(1/Zpv + hess)
      const float Zpost_m = Zpred + Zpost_v * grad;

      llacc += Yt * __logf(p + 1e-10f) + (1.0f - Yt) * __logf(1.0f - p + 1e-10f);

      Zm = Zpost_m; Zv = Zpost_v;
      rS[t] = Zpost_m;   // recycle S tile as Z_filtered staging
      rC[t] = Zpost_v;   // recycle C tile as Z_variance staging
    }
  }
  __syncthreads();

  // ---------- drain outputs: padded LDS tiles -> contiguous global (coalesced) ----------
  if (full) {
#pragma unroll 8
    for (int i = 0; i < PER_THREAD; ++i) {
      const int c   = i * THREADS + tid;
      const size_t goff = (size_t)n0 * TT + (size_t)c * 2;
      const int    loff = (c >> 5) * STRIDE + 2 * (c & 31);
      st_chunk(Zf   + goff, tS + loff);
      st_chunk(Zvar + goff, tC + loff);
    }
  } else {
    for (int i = 0; i < PER_THREAD; ++i) {
      const int c   = i * THREADS + tid;
      const int row = c >> 5;
      if (n0 + row < N) {
        const size_t goff = (size_t)n0 * TT + (size_t)c * 2;
        const int    loff = row * STRIDE + 2 * (c & 31);
        st_chunk(Zf   + goff, tS + loff);
        st_chunk(Zvar + goff, tC + loff);
      }
    }
  }

  // ---------- deterministic block reduction of log-likelihood ----------
  red[tid] = llacc;
  __syncthreads();
  for (int k = THREADS / 2; k > 0; k >>= 1) {
    if (tid < k) red[tid] += red[tid + k];
    __syncthreads();
  }
  if (tid == 0) ws[blockIdx.x] = red[0];

#if USE_ASYNC
  wait_async0();   // async stores drained before wave retires (s_endpgm also waits idle)
#endif
}

__global__ __launch_bounds__(256) void ll_reduce_kernel(const float* __restrict__ part,
                                                        int n, float* __restrict__ out)
{
  __shared__ float s[256];
  float a = 0.0f;
  for (int i = threadIdx.x; i < n; i += 256) a += part[i];   // fixed order -> deterministic
  s[threadIdx.x] = a;
  __syncthreads();
  for (int k = 128; k > 0; k >>= 1) {
    if (threadIdx.x < k) s[threadIdx.x] += s[threadIdx.x + k];
    __syncthreads();
  }
  if (threadIdx.x == 0) out[0] = s[0];
}

extern "C" void kernel_launch(void* const* d_in, const int* in_sizes, int n_in,
                              void* d_out, int out_size, void* d_ws, size_t ws_size,
                              hipStream_t stream) {
  const float* G = (const float*)d_in[0];
  const float* S = (const float*)d_in[1];
  const float* C = (const float*)d_in[2];
  const float* Y = (const float*)d_in[3];
  const float* L = (const float*)d_in[4];

  const int N = in_sizes[0];
  const int T = in_sizes[1] / N;          // == TT (64) per reference
  const int K = in_sizes[4] / N;          // == 3 per reference
  (void)T; (void)n_in; (void)out_size; (void)ws_size;

  float* Zf    = (float*)d_out;
  float* Zvar  = Zf + (size_t)N * TT;
  float* llout = Zvar + (size_t)N * TT;
  float* ws    = (float*)d_ws;

  const int blocks = (N + ROWS - 1) / ROWS;

  hmm_filter_kernel<<<blocks, THREADS, 0, stream>>>(
      G, S, C, Y, L,
      (const float*)d_in[5],  (const float*)d_in[6],  (const float*)d_in[7],
      (const float*)d_in[8],  (const float*)d_in[9],  (const float*)d_in[10],
      (const float*)d_in[11], (const float*)d_in[12], (const float*)d_in[13],
      (const float*)d_in[14], (const float*)d_in[15], (const float*)d_in[16],
      (const float*)d_in[17], (const float*)d_in[18],
      Zf, Zvar, ws, N, K);

  ll_reduce_kernel<<<1, 256, 0, stream>>>(ws, blocks, llout);
}